// LinearCRFLoss_23098334118066
// MI455X (gfx1250) — compile-verified
//
#include <hip/hip_runtime.h>

// LinearCRFLoss for MI455X (gfx1250, wave32).
// B=64, S=2048, O=256 fixed by the reference.

#define S_LEN 2048
#define O_LEN 256
#define B_LEN 64
#define LN256 5.545177444479562f   // ln(256)

typedef float v2f __attribute__((ext_vector_type(2)));
typedef float v4f __attribute__((ext_vector_type(4)));
typedef float v8f __attribute__((ext_vector_type(8)));

__device__ __forceinline__ float wredSum(float v) {
#pragma unroll
  for (int o = 16; o > 0; o >>= 1) v += __shfl_xor(v, o, 32);
  return v;
}
__device__ __forceinline__ float wredMax(float v) {
#pragma unroll
  for (int o = 16; o > 0; o >>= 1) v = fmaxf(v, __shfl_xor(v, o, 32));
  return v;
}
__device__ __forceinline__ int wredSumI(int v) {
#pragma unroll
  for (int o = 16; o > 0; o >>= 1) v += __shfl_xor(v, o, 32);
  return v;
}
__device__ __forceinline__ int wredMaxI(int v) {
#pragma unroll
  for (int o = 16; o > 0; o >>= 1) { int u = __shfl_xor(v, o, 32); v = (u > v) ? u : v; }
  return v;
}

// ---------------- Kernel 1: transition log_softmax stats (per row) -----------
// logZt[i] = logsumexp(transition[i,:]); rowContrib[i] = rowsum(trans_ls[i,:]).
__global__ void crf_trans_kernel(const float* __restrict__ transition,
                                 float* __restrict__ logZt,
                                 float* __restrict__ rowContrib) {
  __shared__ float sm[8];
  __shared__ float bval;
  const int i = blockIdx.x;
  const int t = threadIdx.x;           // 256 threads = 8 waves
  const int lane = t & 31, w = t >> 5;

  const float x = transition[i * O_LEN + t];

  float m = wredMax(x);
  if (lane == 0) sm[w] = m;
  __syncthreads();
  if (t == 0) { float v = sm[0]; for (int j = 1; j < 8; ++j) v = fmaxf(v, sm[j]); bval = v; }
  __syncthreads();
  const float mx = bval;
  __syncthreads();                      // protect sm reuse

  float e = wredSum(__expf(x - mx));
  if (lane == 0) sm[w] = e;
  __syncthreads();
  if (t == 0) { float v = 0.f; for (int j = 0; j < 8; ++j) v += sm[j]; bval = v; }
  __syncthreads();
  const float lz = mx + __logf(bval);
  __syncthreads();                      // protect sm reuse

  float s = wredSum(x);
  if (lane == 0) sm[w] = s;
  __syncthreads();
  if (t == 0) {
    float v = 0.f; for (int j = 0; j < 8; ++j) v += sm[j];
    logZt[i] = lz;
    rowContrib[i] = v - (float)O_LEN * lz;
  }
}

// ---------------- Kernel 2: streaming pass over pred (memory-bound) ----------
// One wave per (b,s) row of 256 floats: 2x nontemporal b128 loads per lane.
__global__ void crf_rows_kernel(const float* __restrict__ pred,
                                const int* __restrict__ gt,
                                float* __restrict__ row_tot,
                                float* __restrict__ emit) {
  const int lane = threadIdx.x & 31;
  const int waveId = (int)((blockIdx.x * blockDim.x + threadIdx.x) >> 5);
  const int totalWaves = (int)((gridDim.x * blockDim.x) >> 5);
  const int rows = B_LEN * S_LEN;

  for (int row = waveId; row < rows; row += totalWaves) {
    const float* base = pred + (size_t)row * O_LEN;
    // prefetch next row this wave will touch (speculative; OOB is dropped)
    __builtin_prefetch(base + (size_t)totalWaves * O_LEN, 0, 0);

    const v4f* p = (const v4f*)base;
    const v4f v0 = __builtin_nontemporal_load(p + lane);
    const v4f v1 = __builtin_nontemporal_load(p + 32 + lane);
    const int g = gt[row];              // uniform across the wave

    float rt, em;
    if (g == 0) {                       // padded: pred row zeroed -> logp = -ln(O)
      rt = -(float)O_LEN * LN256;
      em = -LN256;
    } else {
      float mx = fmaxf(fmaxf(fmaxf(v0.x, v0.y), fmaxf(v0.z, v0.w)),
                       fmaxf(fmaxf(v1.x, v1.y), fmaxf(v1.z, v1.w)));
      mx = wredMax(mx);
      float sx = (v0.x + v0.y) + (v0.z + v0.w) + (v1.x + v1.y) + (v1.z + v1.w);
      sx = wredSum(sx);
      float se = __expf(v0.x - mx) + __expf(v0.y - mx) + __expf(v0.z - mx) + __expf(v0.w - mx)
               + __expf(v1.x - mx) + __expf(v1.y - mx) + __expf(v1.z - mx) + __expf(v1.w - mx);
      se = wredSum(se);
      const float logZ = mx + __logf(se);
      rt = sx - (float)O_LEN * logZ;

      float xg = 0.0f;                  // element at index g (exactly one lane owns it)
      const int j0 = g - 4 * lane;
      if (j0 >= 0 && j0 < 4) xg = v0[j0];
      const int j1 = g - 128 - 4 * lane;
      if (j1 >= 0 && j1 < 4) xg = v1[j1];
      xg = wredSum(xg);
      em = xg - logZ;
    }
    if (lane == 0) { row_tot[row] = rt; emit[row] = em; }
  }
}

// ---------------- Kernel 3: per-batch gold score + pad count -----------------
__global__ void crf_gold_kernel(const float* __restrict__ transition,
                                const float* __restrict__ logZt,
                                const float* __restrict__ emit,
                                const int* __restrict__ gt,
                                float* __restrict__ gold,
                                int* __restrict__ padcount) {
  __shared__ float smf[8];
  __shared__ int smi[8];
  const int b = blockIdx.x;
  const int t = threadIdx.x, lane = t & 31, w = t >> 5;
  const int* gb = gt + b * S_LEN;
  const float* eb = emit + b * S_LEN;

  float acc = 0.f;
  int pc = 0;
  for (int s = t; s < S_LEN; s += 256) {
    const int g1 = gb[s];
    if (g1 == 0) pc++;
    if (s == 0) {
      acc += (float)(S_LEN - 1) * eb[0];       // emit[:,0:1] broadcast over S-1 terms
    } else {
      float tr = 0.f;
      if (g1 != 0) {
        const int g0 = gb[s - 1];
        tr = transition[g0 * O_LEN + g1] - logZt[g0];
      }
      acc += tr + eb[s];
    }
  }
  acc = wredSum(acc);
  pc = wredSumI(pc);
  if (lane == 0) { smf[w] = acc; smi[w] = pc; }
  __syncthreads();
  if (t == 0) {
    float v = 0.f; int c = 0;
    for (int j = 0; j < 8; ++j) { v += smf[j]; c += smi[j]; }
    gold[b] = v;
    padcount[b] = c;
  }
}

// ---------------- Kernel 4: npl -> incl weights, T, transTotal ---------------
__global__ void crf_scalars_kernel(const int* __restrict__ padcount,
                                   const float* __restrict__ rowContrib,
                                   float* __restrict__ incl,
                                   float* __restrict__ scal) {
  __shared__ float smf[8];
  __shared__ int smi[8];
  __shared__ int bmax;
  const int t = threadIdx.x, lane = t & 31, w = t >> 5;  // 256 threads

  int pc = (t < B_LEN) ? padcount[t] : 0;
  pc = wredMaxI(pc);
  if (lane == 0) smi[w] = pc;
  __syncthreads();
  if (t == 0) { int v = smi[0]; for (int j = 1; j < 8; ++j) v = (smi[j] > v) ? smi[j] : v; bmax = v; }
  __syncthreads();

  const int npl = S_LEN - bmax;
  const bool cond = (npl >= 1) && (npl <= S_LEN - 1);
  const int limit = cond ? npl : (S_LEN - 1);

  // incl indexed by k = t-1 (t = 1..S-1); incl[S_LEN-1] is zero-padding for WMMA.
  for (int k = t; k < S_LEN; k += 256)
    incl[k] = ((k + 1) <= limit) ? 1.0f : 0.0f;

  float rc = wredSum(rowContrib[t]);
  if (lane == 0) smf[w] = rc;
  __syncthreads();
  if (t == 0) {
    float v = 0.f; for (int j = 0; j < 8; ++j) v += smf[j];
    scal[0] = (float)limit;   // T = incl.sum()
    scal[1] = v;              // transTotal = sum of all log_softmax(transition)
  }
}

// ---------------- Kernel 5: einsum('bt,t->b') via V_WMMA_F32_16X16X4_F32 -----
// A = incl replicated across 16 rows; B[k][j] = row_tot[b=j_tile][t=k+1].
// D rows are identical; read D[0][j] from c[0] on lanes 0..15.
__global__ void crf_einsum_wmma(const float* __restrict__ row_tot,
                                const float* __restrict__ incl,
                                float* __restrict__ alphaDot) {
  const int lane = threadIdx.x & 31;
  const int wave = threadIdx.x >> 5;   // 0..3 -> batch tiles of 16
  const int col = lane & 15;
  const int half = lane >> 4;          // K half: lanes 0-15 -> K{0,1}, 16-31 -> K{2,3}
  const int b = wave * 16 + col;
  const float* rb = row_tot + b * S_LEN;

  v8f c = {0.f, 0.f, 0.f, 0.f, 0.f, 0.f, 0.f, 0.f};
  for (int k0 = 0; k0 < S_LEN; k0 += 4) {
    const int kA = k0 + half * 2;
    const int kB = kA + 1;
    v2f a, bm;
    a.x = incl[kA];                    // replicated over all A rows
    a.y = incl[kB];
    int tA = kA + 1; if (tA > S_LEN - 1) tA = S_LEN - 1;  // incl[k]=0 there anyway
    int tB = kB + 1; if (tB > S_LEN - 1) tB = S_LEN - 1;
    bm.x = rb[tA];
    bm.y = rb[tB];
    c = __builtin_amdgcn_wmma_f32_16x16x4_f32(false, a, false, bm, (short)0, c,
                                              false, false);
  }
  if (lane < 16) alphaDot[wave * 16 + lane] = c[0];
}

// ---------------- Kernel 6: final deterministic reduction --------------------
__global__ void crf_final_kernel(const float* __restrict__ row_tot,
                                 const float* __restrict__ alphaDot,
                                 const float* __restrict__ gold,
                                 const float* __restrict__ scal,
                                 float* __restrict__ out) {
  __shared__ float sm[64];
  const int t = threadIdx.x;           // 64 threads
  const float T = scal[0];
  const float transTotal = scal[1];
  // sum_o alpha[t,o] - O*gold[t]
  float contrib = row_tot[t * S_LEN]
                + (float)O_LEN * alphaDot[t]
                + T * transTotal
                - (float)O_LEN * gold[t];
  sm[t] = contrib;
  __syncthreads();
  for (int o = 32; o > 0; o >>= 1) {
    if (t < o) sm[t] += sm[t + o];
    __syncthreads();
  }
  if (t == 0) out[0] = sm[0] / (float)(B_LEN * O_LEN);
}

extern "C" void kernel_launch(void* const* d_in, const int* in_sizes, int n_in,
                              void* d_out, int out_size, void* d_ws, size_t ws_size,
                              hipStream_t stream) {
  const float* pred = (const float*)d_in[0];        // (B,S,O) f32
  const int* gt = (const int*)d_in[1];              // (B,S) int
  const float* transition = (const float*)d_in[2];  // (O,O) f32
  float* out = (float*)d_out;

  // Workspace layout (floats): ~1.06 MB total.
  float* ws = (float*)d_ws;
  float* row_tot    = ws;                           // B*S
  float* emit       = row_tot + B_LEN * S_LEN;      // B*S
  float* logZt      = emit + B_LEN * S_LEN;         // O
  float* rowContrib = logZt + O_LEN;                // O
  float* incl       = rowContrib + O_LEN;           // S (last entry = 0 pad)
  float* alphaDot   = incl + S_LEN;                 // B
  float* gold       = alphaDot + B_LEN;             // B
  float* scal       = gold + B_LEN;                 // 8 (T, transTotal)
  int* padcount     = (int*)(scal + 8);             // B

  crf_trans_kernel<<<O_LEN, 256, 0, stream>>>(transition, logZt, rowContrib);
  crf_rows_kernel<<<4096, 256, 0, stream>>>(pred, gt, row_tot, emit);
  crf_gold_kernel<<<B_LEN, 256, 0, stream>>>(transition, logZt, emit, gt, gold, padcount);
  crf_scalars_kernel<<<1, 256, 0, stream>>>(padcount, rowContrib, incl, scal);
  crf_einsum_wmma<<<1, 128, 0, stream>>>(row_tot, incl, alphaDot);
  crf_final_kernel<<<1, 64, 0, stream>>>(row_tot, alphaDot, gold, scal, out);
}